// TemporalGCN_39754217292448
// MI455X (gfx1250) — compile-verified
//
#include <hip/hip_runtime.h>
#include <hip/hip_bf16.h>

typedef __attribute__((ext_vector_type(2))) float v2f;
typedef __attribute__((ext_vector_type(8))) float v8f;

#define CH   64
#define CH2  128

// ---------------- degree / norm ----------------
__global__ __launch_bounds__(256) void k_init_deg(float* deg, int n) {
  int i = blockIdx.x * blockDim.x + threadIdx.x;
  if (i < n) deg[i] = 1.0f;  // self-loop weight 1
}

__global__ __launch_bounds__(256) void k_zero(float* p, size_t n) {
  size_t i = (size_t)blockIdx.x * blockDim.x + threadIdx.x;
  size_t stride = (size_t)gridDim.x * blockDim.x;
  for (; i < n; i += stride) p[i] = 0.0f;
}

__global__ __launch_bounds__(256) void k_deg_accum(const long long* __restrict__ ei,
                                                   const float* __restrict__ w,
                                                   float* __restrict__ deg, int E) {
  int e = blockIdx.x * blockDim.x + threadIdx.x;
  if (e < E) {
    long long d = ei[(size_t)E + e];  // edge_index[1][e]
    atomicAdd(&deg[d], w[e]);
  }
}

__global__ __launch_bounds__(256) void k_rsqrt(float* deg, int n) {
  int i = blockIdx.x * blockDim.x + threadIdx.x;
  if (i < n) deg[i] = rsqrtf(deg[i]);  // deg >= 1 always (self-loop), so valid
}

// ---------------- fp32 WMMA GEMM: one wave per 16x16 tile ----------------
// C[m0:m0+16, c_off+n0 : +16] = A[m0:, a_off:a_off+K] @ B[0:K, n0:n0+16]
__global__ __launch_bounds__(32) void k_gemm16(const float* __restrict__ A, int lda, int a_off,
                                               const float* __restrict__ B, int ldb,
                                               float* __restrict__ C, int ldc, int c_off,
                                               int K) {
  const int m0 = blockIdx.x * 16;
  const int n0 = blockIdx.y * 16;
  const int l  = threadIdx.x;      // 0..31, EXEC all ones
  const int r15 = l & 15;
  const int kh  = (l >> 4) << 1;   // 0 for lanes 0-15, 2 for lanes 16-31

  const float* a = A + (size_t)(m0 + r15) * lda + a_off;  // row = M
  const float* b = B + n0 + r15;                          // col = N, B[k*ldb + n]

  v8f acc = {0.f, 0.f, 0.f, 0.f, 0.f, 0.f, 0.f, 0.f};
  for (int k = 0; k < K; k += 4) {
    v2f av, bv;
    av.x = a[k + kh];
    av.y = a[k + kh + 1];
    bv.x = b[(size_t)(k + kh) * ldb];
    bv.y = b[(size_t)(k + kh + 1) * ldb];
    acc = __builtin_amdgcn_wmma_f32_16x16x4_f32(false, av, false, bv,
                                                (short)0, acc, false, false);
  }
  // C/D layout: VGPR r -> M = r + 8*(lane>=16), N = lane&15
  float* cp = C + (size_t)(m0 + (l >> 4) * 8) * ldc + c_off + n0 + r15;
#pragma unroll
  for (int r = 0; r < 8; ++r) cp[(size_t)r * ldc] = acc[r];
}

// ---------------- edge scatter: msg = xw[src]*norm, atomically into agg[dst] ----------------
__global__ __launch_bounds__(128) void k_scatter(const long long* __restrict__ ei,
                                                 const float* __restrict__ w,
                                                 const float* __restrict__ dinv,
                                                 const float* __restrict__ xw,
                                                 float* __restrict__ agg, int E) {
  int e = blockIdx.x;
  __shared__ float nrm_s;
  __shared__ int s_s, d_s;
  if (threadIdx.x == 0) {
    long long s = ei[e];
    long long d = ei[(size_t)E + e];
    nrm_s = dinv[s] * w[e] * dinv[d];
    s_s = (int)s;
    d_s = (int)d;
  }
  __syncthreads();
  int c = threadIdx.x;  // 0..127 (z-half | h-half)
  atomicAdd(&agg[(size_t)d_s * CH2 + c], xw[(size_t)s_s * CH2 + c] * nrm_s);
}

// ---------------- self-loop + bias, in place into agg ----------------
__global__ __launch_bounds__(256) void k_finalize(float* __restrict__ agg,
                                                  const float* __restrict__ xw,
                                                  const float* __restrict__ dinv,
                                                  const float* __restrict__ bz,
                                                  const float* __restrict__ bh, int n) {
  size_t total = (size_t)n * CH2;
  size_t stride = (size_t)gridDim.x * blockDim.x;
  for (size_t i = (size_t)blockIdx.x * blockDim.x + threadIdx.x; i < total; i += stride) {
    size_t node = i >> 7;
    int c = (int)(i & 127);
    float di = dinv[node];
    float bias = (c < CH) ? bz[c] : bh[c - CH];
    agg[i] += xw[i] * di * di + bias;
  }
}

// ---------------- gates + output head: one wave32 per node ----------------
__global__ __launch_bounds__(256) void k_final(const float* __restrict__ g,
                                               const float* __restrict__ Lzb,
                                               const float* __restrict__ Lhb,
                                               const float* __restrict__ Wlin,
                                               const float* __restrict__ blin,
                                               float* __restrict__ out, int n) {
  int wid = threadIdx.x >> 5;
  int lane = threadIdx.x & 31;
  int node = blockIdx.x * 8 + wid;
  if (node >= n) return;
  float acc = 0.f;
#pragma unroll
  for (int i = 0; i < 2; ++i) {
    int c = lane + 32 * i;
    float zp = g[(size_t)node * CH2 + c] + Lzb[c];
    float hp = g[(size_t)node * CH2 + CH + c] + Lhb[c];
    float z  = 1.f / (1.f + __expf(-zp));
    float ht = tanhf(hp);
    acc += (1.f - z) * ht * Wlin[c];  // H_new = (1-Z)*H_tilde since H==0
  }
#pragma unroll
  for (int off = 16; off > 0; off >>= 1) acc += __shfl_down(acc, off, 32);
  if (lane == 0) out[node] = 1.f / (1.f + __expf(-(acc + blin[0])));
}

// ---------------- launch ----------------
extern "C" void kernel_launch(void* const* d_in, const int* in_sizes, int n_in,
                              void* d_out, int out_size, void* d_ws, size_t ws_size,
                              hipStream_t stream) {
  const float*     x    = (const float*)d_in[0];
  const long long* ei   = (const long long*)d_in[1];  // int64 [2, E]
  const float*     ea   = (const float*)d_in[2];
  const float*     Wz   = (const float*)d_in[3];
  const float*     bz   = (const float*)d_in[4];
  // d_in[5], d_in[6] (Wr, br): dead — reset gate only feeds H*R with H==0
  const float*     Wh   = (const float*)d_in[7];
  const float*     bh   = (const float*)d_in[8];
  const float*     Lzw  = (const float*)d_in[9];
  const float*     Lzb  = (const float*)d_in[10];
  // d_in[11], d_in[12] (Lr): dead
  const float*     Lhw  = (const float*)d_in[13];
  const float*     Lhb  = (const float*)d_in[14];
  const float*     Wlin = (const float*)d_in[15];
  const float*     blin = (const float*)d_in[16];
  float*           out  = (float*)d_out;

  const int N = in_sizes[0] / CH;  // 100000 (multiple of 16)
  const int E = in_sizes[1] / 2;   // 1600000

  // workspace: [deg|dinv : padded N] [xw : N*128] [agg : N*128]  (~103 MB)
  float* wsf = (float*)d_ws;
  float* deg = wsf;
  float* xw  = wsf + 102400;
  float* agg = xw + (size_t)N * CH2;

  // 1) symmetric GCN norm: deg -> dinv (in place)
  k_init_deg<<<(N + 255) / 256, 256, 0, stream>>>(deg, N);
  k_zero<<<2048, 256, 0, stream>>>(agg, (size_t)N * CH2);
  k_deg_accum<<<(E + 255) / 256, 256, 0, stream>>>(ei, ea, deg, E);
  k_rsqrt<<<(N + 255) / 256, 256, 0, stream>>>(deg, N);

  // 2) xw = x @ [Wz | Wh]   (fp32 WMMA)
  dim3 g1(N / 16, CH / 16);
  k_gemm16<<<g1, 32, 0, stream>>>(x, CH, 0, Wz, CH, xw, CH2, 0,  CH);
  k_gemm16<<<g1, 32, 0, stream>>>(x, CH, 0, Wh, CH, xw, CH2, CH, CH);

  // 3) message passing over edges
  k_scatter<<<E, CH2, 0, stream>>>(ei, ea, deg, xw, agg, E);

  // 4) conv = agg + xw*dinv^2 (self-loop) + bias, in place
  k_finalize<<<4096, 256, 0, stream>>>(agg, xw, deg, bz, bh, N);

  // 5) gate pre-activations = [conv_z @ Lz_w[:64] | conv_h @ Lh_w[:64]], reuse xw
  k_gemm16<<<g1, 32, 0, stream>>>(agg, CH2, 0,  Lzw, CH, xw, CH2, 0,  CH);
  k_gemm16<<<g1, 32, 0, stream>>>(agg, CH2, CH, Lhw, CH, xw, CH2, CH, CH);

  // 6) sigmoid/tanh gates + 64->1 head + sigmoid
  k_final<<<(N + 7) / 8, 256, 0, stream>>>(xw, Lzb, Lhb, Wlin, blin, out, N);
}